// PseudoLevinsonDurbinRecursion_87643102642599
// MI455X (gfx1250) — compile-verified
//
#include <hip/hip_runtime.h>
#include <hip/hip_bf16.h>
#include <stdint.h>

// Problem constants (from reference): B=4096 batches, M=256, r-length 257.
#define NBATCH 4096
#define MORD   256
#define RLEN   257
#define WAVES_PER_WG 8
#define THREADS (WAVES_PER_WG * 32)
#define NWG (NBATCH / WAVES_PER_WG)

typedef unsigned int u32;
typedef __attribute__((ext_vector_type(4))) u32 v4u;
typedef __attribute__((ext_vector_type(8))) int v8i;
typedef __attribute__((ext_vector_type(4))) int v4i;

__global__ __launch_bounds__(THREADS)
void levdur_kernel(const float* __restrict__ r, float* __restrict__ out) {
    __shared__ float tile[WAVES_PER_WG * RLEN];

    const int tid  = threadIdx.x;
    const int wave = tid >> 5;
    const int lane = tid & 31;
    const int wg   = blockIdx.x;

    // Zero-init the staging tile (also guarantees the compiler sees stores to
    // `tile`, since the TDM writes it behind the compiler's back).
    for (int i = tid; i < WAVES_PER_WG * RLEN; i += THREADS) tile[i] = 0.0f;
    __syncthreads();

#if __has_builtin(__builtin_amdgcn_tensor_load_to_lds)
    if (wave == 0) {
        // ---- Tensor DMA descriptor (D#): 2D tile of (8 rows x 257 f32) ----
        uint64_t gaddr  = (uint64_t)(uintptr_t)(r + (size_t)wg * WAVES_PER_WG * RLEN);
        u32 ldsaddr = (u32)(uintptr_t)(&tile[0]);   // low 32 bits of generic ptr = LDS byte offset

        v4u g0 = {0u, 0u, 0u, 0u};
        g0.s0 = 1u;                                          // count=1, user descriptor
        g0.s1 = ldsaddr;                                     // lds_addr (bytes)
        g0.s2 = (u32)(gaddr & 0xffffffffu);                  // global_addr[31:0]
        g0.s3 = (u32)((gaddr >> 32) & 0x1ffffffu)            // global_addr[56:32]
              | (2u << 30);                                  // type = 2 ("image")

        v8i g1 = {0, 0, 0, 0, 0, 0, 0, 0};
        g1.s0 = (int)(2u << 16);                             // data_size = 2 (4 bytes)
        g1.s1 = (int)(((u32)RLEN & 0xffffu) << 16);          // tensor_dim0[15:0]
        g1.s2 = (int)((((u32)RLEN >> 16) & 0xffffu)          // tensor_dim0[31:16]
              |       (((u32)NBATCH & 0xffffu) << 16));      // tensor_dim1[15:0]
        g1.s3 = (int)((((u32)NBATCH >> 16) & 0xffffu)        // tensor_dim1[31:16]
              |       (((u32)RLEN & 0xffffu) << 16));        // tile_dim0 = 257
        g1.s4 = (int)((u32)WAVES_PER_WG);                    // tile_dim1 = 8, tile_dim2 = 0
        g1.s5 = (int)((u32)RLEN);                            // tensor_dim0_stride[31:0] = 257
        g1.s6 = 0;                                           // stride hi bits = 0
        g1.s7 = 0;

        v4i g2 = {0, 0, 0, 0};                               // unused (2D tensor)
        v4i g3 = {0, 0, 0, 0};
        v8i g4 = {0, 0, 0, 0, 0, 0, 0, 0};                   // extra operand (6-arg form)

        __builtin_amdgcn_tensor_load_to_lds(g0, g1, g2, g3, g4, 0);
        __builtin_amdgcn_s_wait_tensorcnt(0);                // TENSORcnt == 0
    }
    __syncthreads();
#else
    for (int i = tid; i < WAVES_PER_WG * RLEN; i += THREADS)
        tile[i] = r[(size_t)wg * WAVES_PER_WG * RLEN + i];
    __syncthreads();
#endif

    // ---- Per-wave Levinson-Durbin, fully register resident --------------
    // Layout: element j of a / ar lives at lane = j/8, slot = j%8.
    const float* rw = &tile[wave * RLEN];
    float rr[8];
#pragma unroll
    for (int s = 0; s < 8; ++s) rr[s] = rw[1 + 8 * lane + s];   // r[1 + j]
    float E = rw[0];                                            // r[0]

    float a[8], ar[8];
#pragma unroll
    for (int s = 0; s < 8; ++s) { a[s] = 0.0f; ar[s] = 0.0f; }

#pragma unroll 1
    for (int mh = 0; mh < 32; ++mh) {           // m = 8*mh + s
#pragma unroll
        for (int s = 0; s < 8; ++s) {
            // acc = r[m+1] + sum_{i<m} a_i * r[m-i]  ==  r[m+1] + sum_j ar_j * r[j+1]
            float acc = __shfl(rr[s], mh, 32);              // r[m+1] (uniform lane)
#pragma unroll
            for (int q = 0; q < 8; ++q) acc = fmaf(ar[q], rr[q], acc);
            // wave32 all-lanes reduction
#pragma unroll
            for (int mask = 16; mask >= 1; mask >>= 1)
                acc += __shfl_xor(acc, mask, 32);

            // k = -acc / E   (fast rcp + one Newton step)
            float re = __builtin_amdgcn_rcpf(E);
            re = re * (2.0f - E * re);
            float k = -acc * re;
            E = fmaf(-k * k, E, E);                          // E *= (1 - k^2)

            // t = ar + k*a  (old values), then ar' = shift-by-1(t), prepend k
            float t7 = fmaf(k, a[7], ar[7]);
            float carry = __shfl_up(t7, 1, 32);              // t[8*(lane-1)+7]
            float tq[7];
#pragma unroll
            for (int q = 0; q < 7; ++q) tq[q] = fmaf(k, a[q], ar[q]);
            // a' = a + k*ar (old ar), then a'[m] = k
#pragma unroll
            for (int q = 0; q < 8; ++q) a[q] = fmaf(k, ar[q], a[q]);
            if (lane == mh) a[s] = k;
#pragma unroll
            for (int q = 7; q >= 1; --q) ar[q] = tq[q - 1];
            ar[0] = (lane == 0) ? k : carry;
        }
    }

    float K = sqrtf(E);    // E_M = r0 + r1.a  =>  K

    // ---- Stage result into LDS, then fully coalesced store --------------
    __syncthreads();       // done reading tile
    float* ow = &tile[wave * RLEN];
    if (lane == 0) ow[0] = K;
#pragma unroll
    for (int s = 0; s < 8; ++s) ow[1 + 8 * lane + s] = a[s];
    __syncthreads();

    const size_t obase = (size_t)wg * WAVES_PER_WG * RLEN;
    for (int i = tid; i < WAVES_PER_WG * RLEN; i += THREADS)
        out[obase + i] = tile[i];
}

extern "C" void kernel_launch(void* const* d_in, const int* in_sizes, int n_in,
                              void* d_out, int out_size, void* d_ws, size_t ws_size,
                              hipStream_t stream) {
    (void)in_sizes; (void)n_in; (void)d_ws; (void)ws_size; (void)out_size;
    const float* r = (const float*)d_in[0];
    float* out = (float*)d_out;
    hipLaunchKernelGGL(levdur_kernel, dim3(NWG), dim3(THREADS), 0, stream, r, out);
}